// M_20418274525529
// MI455X (gfx1250) — compile-verified
//
#include <hip/hip_runtime.h>
#include <hip/hip_bf16.h>
#include <math.h>

// ---------------------------------------------------------------------------
// Model dimensions (compile-time constants from the reference)
// ---------------------------------------------------------------------------
#define BB   32
#define LL   256
#define PD   164
#define DD   240
#define HH   16
#define NB   6
#define NC   250
#define DK   15          // DD / HH
#define HID  480         // 2 * DD
#define LP   257         // LL + 1
#define ATT_SCALE 0.2581988897471611f   // 1/sqrt(15)

// GEMM tiling: block = 64 threads (2 waves). Each wave computes a 32x48 C
// block (2 M-tiles x 3 N-tiles of 16x16). Block covers 64 rows x 48 cols.
// M must be a multiple of 64 (we pad activation buffers to 8256 rows),
// N a multiple of 48 (240, 480 both are), K a multiple of 4.
#define KCHUNK 240                       // B-panel rows staged in LDS per step
#define R1P  8256                        // padded row count for [B,Lp] bufs

typedef __attribute__((ext_vector_type(2))) float v2f;
typedef __attribute__((ext_vector_type(8))) float v8f;
typedef __attribute__((ext_vector_type(4))) unsigned int u32x4;
typedef __attribute__((ext_vector_type(8))) int i32x8;
typedef __attribute__((ext_vector_type(4))) int i32x4;

// ---------------------------------------------------------------------------
// wave32 reductions
// ---------------------------------------------------------------------------
__device__ __forceinline__ float wred_sum(float x) {
  #pragma unroll
  for (int m = 16; m > 0; m >>= 1) x += __shfl_xor(x, m, 32);
  return x;
}
__device__ __forceinline__ float wred_max(float x) {
  #pragma unroll
  for (int m = 16; m > 0; m >>= 1) x = fmaxf(x, __shfl_xor(x, m, 32));
  return x;
}

// ---------------------------------------------------------------------------
// TDM: stage a [rows x 48] fp32 panel of W (row stride N) into LDS via the
// Tensor Data Mover.  Issued by one wave; caller barriers afterwards.
// Descriptor layout per CDNA5 ISA ch.8 (D# groups 0/1; groups 2/3 zero for 2D).
// ---------------------------------------------------------------------------
__device__ __forceinline__ void tdm_load_panel(const float* W, int N,
                                               int kc, int rows, int nbase,
                                               float* ldsDst)
{
  const unsigned lds_off = (unsigned)(unsigned long long)(size_t)ldsDst;
  const unsigned long long ga =
      (unsigned long long)(size_t)(W + (size_t)kc * N + nbase);

  u32x4 g0;
  g0.x = 1u;                                        // count=1 (valid), user mode
  g0.y = lds_off;                                   // lds_addr
  g0.z = (unsigned)(ga & 0xFFFFFFFFu);              // global_addr[31:0]
  g0.w = (unsigned)((ga >> 32) & 0x01FFFFFFu)       // global_addr[56:32]
         | (2u << 30);                              // type = 2 ("image")

  i32x8 g1;
  g1[0] = (int)(2u << 16);                          // data_size = 4 bytes
  g1[1] = (int)(((unsigned)N & 0xFFFFu) << 16);     // tensor_dim0[15:0]
  g1[2] = (int)(((unsigned)rows + (unsigned)kc) << 16); // dim0 hi=0 | tensor_dim1 lo
  g1[3] = (int)(48u << 16);                         // tensor_dim1 hi=0 | tile_dim0=48
  g1[4] = (int)((unsigned)rows & 0xFFFFu);          // tile_dim1 = rows (tile_dim2=0)
  g1[5] = (int)N;                                   // tensor_dim0_stride[31:0]
  g1[6] = 0;                                        // stride hi / dim1_stride lo
  g1[7] = 0;

  i32x4 g2 = {0, 0, 0, 0};
  i32x4 g3 = {0, 0, 0, 0};
#if defined(__clang_major__) && (__clang_major__ >= 23)
  i32x8 g4 = {0, 0, 0, 0, 0, 0, 0, 0};
  __builtin_amdgcn_tensor_load_to_lds(g0, g1, g2, g3, g4, 0);
#else
  __builtin_amdgcn_tensor_load_to_lds(g0, g1, g2, g3, 0);
#endif
  __builtin_amdgcn_s_wait_tensorcnt(0);
}

// ---------------------------------------------------------------------------
// fp32 WMMA GEMM:  C[M,N] (= or +=) A[M,K] @ W[K,N] + bias
//   FLAGS (compile-time) bit0: relu    bit1: accumulate into C (residual)
// grid = (N/48, M/64), block = 64 (2 waves). Each wave: 2 M-tiles x 3 N-tiles.
// B panel staged in LDS in KCHUNK-row chunks by the Tensor Data Mover.
// ---------------------------------------------------------------------------
template <int FLAGS>
__global__ __launch_bounds__(64) void gemm_wmma_f32_kernel(
    const float* __restrict__ A, const float* __restrict__ W,
    const float* __restrict__ bias, float* __restrict__ C,
    int M, int N, int K)
{
  __shared__ float Bs[KCHUNK * 48];

  const int tid   = threadIdx.x;
  const int lane  = tid & 31;
  const int wave  = tid >> 5;
  const int nbase = blockIdx.x * 48;
  const int n     = lane & 15;
  const int hi    = lane >> 4;                 // 0: lanes 0-15, 1: lanes 16-31
  const int mbase = blockIdx.y * 64 + wave * 32;

  const float* Arow0 = A + (size_t)(mbase + n) * K;        // M-tile 0 row
  const float* Arow1 = A + (size_t)(mbase + 16 + n) * K;   // M-tile 1 row
  const float* bp    = Bs + 48 * 2 * hi + n;               // LDS frag base

  v8f c[2][3];
  #pragma unroll
  for (int mi = 0; mi < 2; ++mi)
    #pragma unroll
    for (int ni = 0; ni < 3; ++ni) c[mi][ni] = (v8f){};

  for (int kc = 0; kc < K; kc += KCHUNK) {
    const int rows = (K - kc) < KCHUNK ? (K - kc) : KCHUNK;
    if (wave == 0) tdm_load_panel(W, N, kc, rows, nbase, Bs);
    __syncthreads();

    for (int k0 = 0; k0 < rows; k0 += 4) {
      // A fragments (16x4): VGPR0 = K{0|2}, VGPR1 = K{1|3}, hi selects pair
      v2f a0, a1;
      a0.x = Arow0[kc + k0 + 2 * hi];
      a0.y = Arow0[kc + k0 + 2 * hi + 1];
      a1.x = Arow1[kc + k0 + 2 * hi];
      a1.y = Arow1[kc + k0 + 2 * hi + 1];
      // B fragments (4x16) from LDS panel [rows x 48]
      v2f b[3];
      #pragma unroll
      for (int ni = 0; ni < 3; ++ni) {
        b[ni].x = bp[k0 * 48 + ni * 16];
        b[ni].y = bp[(k0 + 1) * 48 + ni * 16];
      }
      #pragma unroll
      for (int ni = 0; ni < 3; ++ni) {
        c[0][ni] = __builtin_amdgcn_wmma_f32_16x16x4_f32(
            false, a0, false, b[ni], (short)0, c[0][ni], false, false);
        c[1][ni] = __builtin_amdgcn_wmma_f32_16x16x4_f32(
            false, a1, false, b[ni], (short)0, c[1][ni], false, false);
      }
    }
    __syncthreads();
  }

  // Epilogue: C/D tile layout -> VGPR j holds M = j (lanes 0-15) / j+8 (16-31)
  #pragma unroll
  for (int ni = 0; ni < 3; ++ni) {
    const int col = nbase + ni * 16 + n;
    const float bv = bias[col];
    #pragma unroll
    for (int mi = 0; mi < 2; ++mi) {
      #pragma unroll
      for (int j = 0; j < 8; ++j) {
        const int row = mbase + mi * 16 + j + 8 * hi;
        float val = c[mi][ni][j] + bv;
        if (FLAGS & 1) val = fmaxf(val, 0.0f);
        const size_t oi = (size_t)row * N + col;
        if (FLAGS & 2) C[oi] += val;
        else           C[oi]  = val;
      }
    }
  }
}

// ---------------------------------------------------------------------------
// Padding mask:  mask[b][0]=0 (CLS);  mask[b][1+l] = (max_f x[b,l,f] == 0)
// ---------------------------------------------------------------------------
__global__ __launch_bounds__(32) void pad_mask_kernel(
    const float* __restrict__ x, float* __restrict__ mask)
{
  const int l = blockIdx.x, b = blockIdx.y, lane = threadIdx.x;
  const float* row = x + ((size_t)b * LL + l) * PD;
  float m = -INFINITY;
  for (int i = lane; i < PD; i += 32) m = fmaxf(m, row[i]);
  m = wred_max(m);
  if (lane == 0) {
    mask[(size_t)b * LP + 1 + l] = (m == 0.0f) ? 1.0f : 0.0f;
    if (l == 0) mask[(size_t)b * LP] = 0.0f;
  }
}

// ---------------------------------------------------------------------------
// LayerNorm over last dim (width Wd), optional fused ReLU. One wave per row.
// ---------------------------------------------------------------------------
__global__ __launch_bounds__(32) void layernorm_kernel(
    const float* __restrict__ src, float* __restrict__ dst,
    const float* __restrict__ g, const float* __restrict__ bta,
    int Wd, int relu)
{
  const int row  = blockIdx.x;
  const int lane = threadIdx.x;
  const float* in = src + (size_t)row * Wd;
  float s1 = 0.0f, s2 = 0.0f;
  for (int i = lane; i < Wd; i += 32) {
    const float v = in[i];
    s1 += v; s2 += v * v;
  }
  s1 = wred_sum(s1); s2 = wred_sum(s2);
  const float mean = s1 / (float)Wd;
  const float var  = s2 / (float)Wd - mean * mean;
  const float rstd = rsqrtf(var + 1e-5f);
  float* out = dst + (size_t)row * Wd;
  for (int i = lane; i < Wd; i += 32) {
    float v = (in[i] - mean) * rstd * g[i] + bta[i];
    if (relu) v = fmaxf(v, 0.0f);
    out[i] = v;
  }
}

// ---------------------------------------------------------------------------
// Assemble h[b, 0] = cls_embed ; h[b, 1+l] = t2[b,l] + pos_embed[l]
// ---------------------------------------------------------------------------
__global__ void assemble_kernel(
    const float* __restrict__ t2, const float* __restrict__ pos,
    const float* __restrict__ cls, float* __restrict__ h)
{
  const size_t idx = (size_t)blockIdx.x * blockDim.x + threadIdx.x;
  const size_t total = (size_t)BB * LP * DD;
  if (idx >= total) return;
  const int d = (int)(idx % DD);
  const int t = (int)((idx / DD) % LP);
  const int b = (int)(idx / ((size_t)DD * LP));
  float v;
  if (t == 0) v = cls[d];
  else        v = t2[((size_t)b * LL + (t - 1)) * DD + d] + pos[(size_t)(t - 1) * DD + d];
  h[idx] = v;
}

// ---------------------------------------------------------------------------
// Attention: one wave per (b, h, q-row). Two-pass softmax with per-lane
// score stash (ceil(257/32) = 9 slots), masked positions get -10000.
// ---------------------------------------------------------------------------
__global__ __launch_bounds__(32) void attention_kernel(
    const float* __restrict__ q, const float* __restrict__ k,
    const float* __restrict__ v, const float* __restrict__ mask,
    float* __restrict__ o)
{
  const int qi = blockIdx.x, hh = blockIdx.y, b = blockIdx.z;
  const int lane = threadIdx.x;

  const float* qrow = q + ((size_t)b * LP + qi) * DD + hh * DK;
  float qv[DK];
  #pragma unroll
  for (int t = 0; t < DK; ++t) qv[t] = qrow[t];

  float s[9];
  float mx = -INFINITY;
  {
    int j = 0;
    for (int kk = lane; kk < LP; kk += 32, ++j) {
      const float* krow = k + ((size_t)b * LP + kk) * DD + hh * DK;
      float d = 0.0f;
      #pragma unroll
      for (int t = 0; t < DK; ++t) d += qv[t] * krow[t];
      d *= ATT_SCALE;
      if (mask[(size_t)b * LP + kk] != 0.0f) d = -10000.0f;
      s[j] = d;
      mx = fmaxf(mx, d);
    }
  }
  mx = wred_max(mx);

  float acc[DK];
  #pragma unroll
  for (int t = 0; t < DK; ++t) acc[t] = 0.0f;
  float sum = 0.0f;
  {
    int j = 0;
    for (int kk = lane; kk < LP; kk += 32, ++j) {
      const float p = expf(s[j] - mx);
      sum += p;
      const float* vrow = v + ((size_t)b * LP + kk) * DD + hh * DK;
      #pragma unroll
      for (int t = 0; t < DK; ++t) acc[t] += p * vrow[t];
    }
  }
  sum = wred_sum(sum);
  #pragma unroll
  for (int t = 0; t < DK; ++t) acc[t] = wred_sum(acc[t]);

  const float inv = 1.0f / sum;
  float outv = 0.0f;
  #pragma unroll
  for (int t = 0; t < DK; ++t) if (lane == t) outv = acc[t];
  if (lane < DK)
    o[((size_t)b * LP + qi) * DD + hh * DK + lane] = outv * inv;
}

// ---------------------------------------------------------------------------
// Masked mean pool + logits
// ---------------------------------------------------------------------------
__global__ __launch_bounds__(256) void pool_kernel(
    const float* __restrict__ h, const float* __restrict__ mask,
    float* __restrict__ pooled)
{
  const int b = blockIdx.x, d = threadIdx.x;
  if (d >= DD) return;
  float s = 0.0f, cnt = 0.0f;
  for (int t = 0; t < LP; ++t) {
    const float w = 1.0f - mask[(size_t)b * LP + t];
    s   += w * h[((size_t)b * LP + t) * DD + d];
    cnt += w;
  }
  pooled[(size_t)b * DD + d] = s / cnt;
}

__global__ __launch_bounds__(256) void logits_kernel(
    const float* __restrict__ pooled, const float* __restrict__ lw,
    const float* __restrict__ lb, float* __restrict__ out)
{
  const int b = blockIdx.x, c = threadIdx.x;
  if (c >= NC) return;
  float s = lb[c];
  for (int d = 0; d < DD; ++d) s += pooled[(size_t)b * DD + d] * lw[(size_t)d * NC + c];
  out[(size_t)b * NC + c] = s;
}

// ---------------------------------------------------------------------------
// Host side
// ---------------------------------------------------------------------------
static inline void gemm(const float* A, const float* W, const float* bias,
                        float* C, int M, int N, int K, int flags,
                        hipStream_t s)
{
  dim3 grid(N / 48, M / 64);
  switch (flags) {
    case 1:  gemm_wmma_f32_kernel<1><<<grid, 64, 0, s>>>(A, W, bias, C, M, N, K); break;
    case 2:  gemm_wmma_f32_kernel<2><<<grid, 64, 0, s>>>(A, W, bias, C, M, N, K); break;
    default: gemm_wmma_f32_kernel<0><<<grid, 64, 0, s>>>(A, W, bias, C, M, N, K); break;
  }
}

extern "C" void kernel_launch(void* const* d_in, const int* in_sizes, int n_in,
                              void* d_out, int out_size, void* d_ws, size_t ws_size,
                              hipStream_t stream) {
  (void)in_sizes; (void)n_in; (void)out_size; (void)ws_size;

  const float* x      = (const float*)d_in[0];
  const float* pos    = (const float*)d_in[1];
  const float* cls    = (const float*)d_in[2];
  const float* e_W1   = (const float*)d_in[3];
  const float* e_b1   = (const float*)d_in[4];
  const float* e_g1   = (const float*)d_in[5];
  const float* e_bn1  = (const float*)d_in[6];
  const float* e_W2   = (const float*)d_in[7];
  const float* e_b2   = (const float*)d_in[8];
  const float* e_g2   = (const float*)d_in[9];
  const float* e_bn2  = (const float*)d_in[10];
  const float* ln1_g  = (const float*)d_in[11];
  const float* ln1_b  = (const float*)d_in[12];
  const float* Wq     = (const float*)d_in[13];
  const float* bq     = (const float*)d_in[14];
  const float* Wk     = (const float*)d_in[15];
  const float* bk     = (const float*)d_in[16];
  const float* Wv     = (const float*)d_in[17];
  const float* bv     = (const float*)d_in[18];
  const float* Wo     = (const float*)d_in[19];
  const float* bo     = (const float*)d_in[20];
  const float* ln2_g  = (const float*)d_in[21];
  const float* ln2_b  = (const float*)d_in[22];
  const float* W1     = (const float*)d_in[23];
  const float* b1     = (const float*)d_in[24];
  const float* W2     = (const float*)d_in[25];
  const float* b2     = (const float*)d_in[26];
  const float* lw     = (const float*)d_in[27];
  const float* lb     = (const float*)d_in[28];
  float* out = (float*)d_out;

  // ---- workspace carve-up (floats); activation buffers padded to R1P rows
  // so GEMM M-tiles (64-row blocks) divide evenly; tail rows hold garbage
  // that is written/read only by the GEMM padding lanes and never consumed.
  const size_t R0 = (size_t)BB * LL;   // 8192 (= 128 * 64, no padding needed)
  const size_t R1 = (size_t)BB * LP;   // 8224 rows actually used
  float* ws = (float*)d_ws;
  size_t off = 0;
  float* mask   = ws + off; off += R1;
  float* buf480 = ws + off; off += (size_t)R1P * HID;   // t1 / FFN hidden
  float* h      = ws + off; off += (size_t)R1P * DD;
  float* y      = ws + off; off += (size_t)R1P * DD;
  float* q      = ws + off; off += (size_t)R1P * DD;    // also t2 pre-loop
  float* kbuf   = ws + off; off += (size_t)R1P * DD;
  float* vbuf   = ws + off; off += (size_t)R1P * DD;
  float* attno  = ws + off; off += (size_t)R1P * DD;
  float* pooled = ws + off; off += (size_t)BB * DD;

  // ---- padding mask ----
  pad_mask_kernel<<<dim3(LL, BB), 32, 0, stream>>>(x, mask);

  // ---- XEmbed ----
  gemm(x, e_W1, e_b1, buf480, (int)R0, HID, PD, 0, stream);       // t1
  layernorm_kernel<<<(int)R0, 32, 0, stream>>>(buf480, buf480, e_g1, e_bn1, HID, 1);
  gemm(buf480, e_W2, e_b2, q, (int)R0, DD, HID, 0, stream);       // t2 (in q)
  layernorm_kernel<<<(int)R0, 32, 0, stream>>>(q, q, e_g2, e_bn2, DD, 1);

  // ---- h = [cls ; t2 + pos] ----
  {
    const size_t total = R1 * DD;
    assemble_kernel<<<(int)((total + 255) / 256), 256, 0, stream>>>(q, pos, cls, h);
  }

  // ---- transformer blocks ----
  for (int i = 0; i < NB; ++i) {
    const float* wq = Wq + (size_t)i * DD * DD;
    const float* wk = Wk + (size_t)i * DD * DD;
    const float* wv = Wv + (size_t)i * DD * DD;
    const float* wo = Wo + (size_t)i * DD * DD;
    const float* w1 = W1 + (size_t)i * DD * HID;
    const float* w2 = W2 + (size_t)i * HID * DD;

    // pre-norm attention
    layernorm_kernel<<<(int)R1, 32, 0, stream>>>(h, y, ln1_g + (size_t)i * DD,
                                                 ln1_b + (size_t)i * DD, DD, 0);
    gemm(y, wq, bq + (size_t)i * DD, q,    R1P, DD, DD, 0, stream);
    gemm(y, wk, bk + (size_t)i * DD, kbuf, R1P, DD, DD, 0, stream);
    gemm(y, wv, bv + (size_t)i * DD, vbuf, R1P, DD, DD, 0, stream);
    attention_kernel<<<dim3(LP, HH, BB), 32, 0, stream>>>(q, kbuf, vbuf, mask, attno);
    gemm(attno, wo, bo + (size_t)i * DD, h, R1P, DD, DD, 2, stream);  // h += ...

    // pre-norm FFN
    layernorm_kernel<<<(int)R1, 32, 0, stream>>>(h, y, ln2_g + (size_t)i * DD,
                                                 ln2_b + (size_t)i * DD, DD, 0);
    gemm(y, w1, b1 + (size_t)i * HID, buf480, R1P, HID, DD, 1, stream);  // relu
    gemm(buf480, w2, b2 + (size_t)i * DD, h,  R1P, DD, HID, 2, stream);  // h += ...
  }

  // ---- masked mean pool + logits ----
  pool_kernel<<<BB, 256, 0, stream>>>(h, mask, pooled);
  logits_kernel<<<BB, 256, 0, stream>>>(pooled, lw, lb, out);
}